// CustomLoss_82257213653439
// MI455X (gfx1250) — compile-verified
//
#include <hip/hip_runtime.h>

// ---------------------------------------------------------------------------
// Weighted NLL: mean_i( -w_i * log(outputs[i, targets[i]]) ),
// w_i = targets[i] != 0 ? 0.7 : 0.3.   N = 1048576, C = 128.
//
// MI455X (gfx1250, wave32) design, from the roofline:
//   traffic = N*128B gather lines (134 MB) + 4 MB targets @ 23.3 TB/s ~ 6 us;
//   compute (1M v_log_f32) is negligible -> pure scattered-load bandwidth.
//  - 8 samples/thread: 2x b128 coalesced target loads, then 8 scattered
//    non-temporal b32 gathers issued back-to-back (8 loads in flight/lane).
//  - Per-wave reduction of the 64 accumulators (2 per lane) via a single
//    V_WMMA_F32_16X16X4_F32 (A = ones 16x4, B = accumulators 4x16), then
//    4 ds_swizzle_b32 SWAPX steps (no bpermute clamp overhead).
//  - Deterministic two-pass tree, no float atomics; graph-capture safe.
// ---------------------------------------------------------------------------

typedef __attribute__((ext_vector_type(2))) float v2f;
typedef __attribute__((ext_vector_type(8))) float v8f;
typedef __attribute__((ext_vector_type(4))) int   v4i;

#define C_LOG2   7          // C = 128
#define THREADS  256
#define SPT      8          // samples per thread

// ds_swizzle group-of-32 encoding: offset[14:10]=xor, [9:5]=or, [4:0]=and.
#define SWZ_XOR(m)  (((m) << 10) | 0x1F)

template <int PAT>
__device__ __forceinline__ float swz_xor_add(float x) {
  int y = __builtin_amdgcn_ds_swizzle(__float_as_int(x), PAT);
  return x + __int_as_float(y);
}

// Sum the 64 f32 accumulators held by one wave32 (2 per lane) using the
// matrix unit:  D = ones(16x4) x B(4x16) -> D[m][n] = colsum(n) for all m.
// D's first VGPR holds colsum[lane&15] in every lane (lanes 16-31 carry the
// M=8 row, identical values); 4 in-half xor-swizzle adds finish the total.
__device__ __forceinline__ float wave_sum64_wmma(float acc0, float acc1) {
  v2f a; a[0] = 1.0f; a[1] = 1.0f;     // A: 16x4 all ones
  v2f b; b[0] = acc0; b[1] = acc1;     // B: this wave's 64 partials
  v8f c = {};
  v8f d = __builtin_amdgcn_wmma_f32_16x16x4_f32(
      /*neg_a=*/false, a, /*neg_b=*/false, b,
      /*c_mod=*/(short)0, c, /*reuse_a=*/false, /*reuse_b=*/false);
  float s = d[0];
  s = swz_xor_add<SWZ_XOR(1)>(s);
  s = swz_xor_add<SWZ_XOR(2)>(s);
  s = swz_xor_add<SWZ_XOR(4)>(s);
  s = swz_xor_add<SWZ_XOR(8)>(s);
  return s;                             // every lane: wave total of 64 values
}

// Block total, valid in threadIdx.x == 0. Fixed order -> deterministic.
__device__ __forceinline__ float block_sum(float acc0, float acc1) {
  __shared__ float lds[THREADS / 32];
  float w = wave_sum64_wmma(acc0, acc1);
  if ((threadIdx.x & 31) == 0) lds[threadIdx.x >> 5] = w;
  __syncthreads();
  float total = 0.0f;
  if (threadIdx.x == 0) {
#pragma unroll
    for (int i = 0; i < THREADS / 32; ++i) total += lds[i];
  }
  return total;
}

__global__ void __launch_bounds__(THREADS)
nll_partial(const float* __restrict__ outputs, const int* __restrict__ targets,
            float* __restrict__ partial, int n) {
  const long base = ((long)blockIdx.x * THREADS + threadIdx.x) * SPT;
  float acc0 = 0.0f, acc1 = 0.0f;

  if (base + SPT <= (long)n) {
    // Coalesced, streaming target loads (32B-aligned: base % 8 == 0).
    v4i t0 = __builtin_nontemporal_load((const v4i*)(targets + base));
    v4i t1 = __builtin_nontemporal_load((const v4i*)(targets + base + 4));
    int t[SPT] = { t0[0], t0[1], t0[2], t0[3], t1[0], t1[1], t1[2], t1[3] };

    // Issue all 8 scattered gathers before consuming any (max loads in
    // flight). Non-temporal: every cacheline is touched exactly once.
    float p[SPT];
#pragma unroll
    for (int k = 0; k < SPT; ++k)
      p[k] = __builtin_nontemporal_load(
          outputs + (((size_t)(base + k)) << C_LOG2) + t[k]);

#pragma unroll
    for (int k = 0; k < SPT; ++k) {
      float w    = (t[k] != 0) ? 0.7f : 0.3f;
      float term = w * __logf(p[k]);
      if (k & 1) acc1 += term; else acc0 += term;
    }
  } else {
    // Tail (never taken for N = 2^20, kept for generality).
    for (long i = base; i < (long)n; ++i) {
      int   t = targets[i];
      float w = (t != 0) ? 0.7f : 0.3f;
      acc0 += w * __logf(outputs[((size_t)i << C_LOG2) + t]);
    }
  }

  float total = block_sum(acc0, acc1);
  if (threadIdx.x == 0) partial[blockIdx.x] = total;
}

__global__ void __launch_bounds__(THREADS)
nll_finalize(const float* __restrict__ partial, int nparts,
             float* __restrict__ out, float inv_n) {
  float acc0 = 0.0f, acc1 = 0.0f;
  for (int i = threadIdx.x; i < nparts; i += 2 * THREADS)           acc0 += partial[i];
  for (int i = threadIdx.x + THREADS; i < nparts; i += 2 * THREADS) acc1 += partial[i];

  float total = block_sum(acc0, acc1);
  if (threadIdx.x == 0) out[0] = -total * inv_n;   // mean of -w*log(p)
}

extern "C" void kernel_launch(void* const* d_in, const int* in_sizes, int n_in,
                              void* d_out, int out_size, void* d_ws, size_t ws_size,
                              hipStream_t stream) {
  const float* outputs = (const float*)d_in[0];   // [N, 128] f32
  const int*   targets = (const int*)d_in[1];     // [N] int32
  float*       out     = (float*)d_out;           // [1] f32
  float*       partial = (float*)d_ws;            // block partials

  const int n      = in_sizes[1];                       // N = 1048576
  const int blocks = (n + THREADS * SPT - 1) / (THREADS * SPT);  // 512

  nll_partial<<<blocks, THREADS, 0, stream>>>(outputs, targets, partial, n);
  nll_finalize<<<1, THREADS, 0, stream>>>(partial, blocks, out, 1.0f / (float)n);
}